// GETD_HT3_43284680409616
// MI455X (gfx1250) — compile-verified
//
#include <hip/hip_runtime.h>
#include <hip/hip_bf16.h>
#include <math.h>

typedef __attribute__((ext_vector_type(2))) float v2f;
typedef __attribute__((ext_vector_type(8))) float v8f;

#ifndef __has_builtin
#define __has_builtin(x) 0
#endif
#if __has_builtin(__builtin_amdgcn_wmma_f32_16x16x4_f32)
#define HAVE_WMMA_F32 1
#else
#define HAVE_WMMA_F32 0
#endif

#define BATCH 128
#define DE_ 32
#define DR_ 32
#define R1_ 40
#define R2_ 40
#define R3_ 40
#define NENT 100000
#define EPS_ 1e-5f
#define PRED_ELEMS ((size_t)BATCH * NENT)

// ---------------- K1: gather + BatchNorm (training-mode batch stats) --------
// block x: 0 = relation (R_weight, gamma_r/beta_r), 1..3 = e2/e3/e4 (E_weight)
__global__ void bn_gather_kernel(const float* __restrict__ Rw,
                                 const float* __restrict__ Ew,
                                 const int* __restrict__ r_idx,
                                 const int* __restrict__ e2_idx,
                                 const int* __restrict__ e3_idx,
                                 const int* __restrict__ e4_idx,
                                 const float* __restrict__ gr, const float* __restrict__ br,
                                 const float* __restrict__ ge, const float* __restrict__ be,
                                 float* __restrict__ out) {
  int which = blockIdx.x;
  int col = threadIdx.x;               // 0..31
  const int* idx;
  const float* tab; const float* gamma; const float* beta;
  if (which == 0)      { idx = r_idx;  tab = Rw; gamma = gr; beta = br; }
  else if (which == 1) { idx = e2_idx; tab = Ew; gamma = ge; beta = be; }
  else if (which == 2) { idx = e3_idx; tab = Ew; gamma = ge; beta = be; }
  else                 { idx = e4_idx; tab = Ew; gamma = ge; beta = be; }
  float s = 0.f, ss = 0.f;
  for (int b = 0; b < BATCH; ++b) {
    float x = tab[(size_t)idx[b] * 32 + col];
    s += x; ss += x * x;
  }
  float m = s / BATCH;
  float v = ss / BATCH - m * m;
  float inv = rsqrtf(v + EPS_);
  float g = gamma[col], bb = beta[col];
  float* dst = out + (size_t)which * BATCH * 32;
  for (int b = 0; b < BATCH; ++b) {
    float x = tab[(size_t)idx[b] * 32 + col];
    dst[b * 32 + col] = (x - m) * inv * g + bb;
  }
}

// ---------------- K2: rroot[b,c] = r_emb[b,:] @ root[:,c] -------------------
__global__ void rroot_kernel(const float* __restrict__ remb,
                             const float* __restrict__ root,
                             float* __restrict__ rroot) {
  int idx = blockIdx.x * blockDim.x + threadIdx.x;
  if (idx >= BATCH * R3_) return;
  int b = idx / R3_, c = idx % R3_;
  float s = 0.f;
  for (int d = 0; d < DR_; ++d) s += remb[b * DR_ + d] * root[d * R3_ + c];
  rroot[idx] = s;
}

// ---------------- K3: K3[d,a,b2] = sum_c root[d,c] * internal[c,a,b2] -------
__global__ void k3_kernel(const float* __restrict__ root,
                          const float* __restrict__ internal,
                          float* __restrict__ k3) {
  int idx = blockIdx.x * blockDim.x + threadIdx.x;
  if (idx >= DR_ * R1_ * R2_) return;
  int d = idx / (R1_ * R2_);
  int rem = idx % (R1_ * R2_);
  int a = rem / R2_, b2 = rem % R2_;
  float s = 0.f;
  for (int c = 0; c < R3_; ++c) s += root[d * R3_ + c] * internal[(c * R1_ + a) * R2_ + b2];
  k3[idx] = s;
}

// ---------------- K4: P[d,b2,ij] = sum_a K3[d,a,b2] * left[a,ij] ------------
__global__ void p_kernel(const float* __restrict__ k3,
                         const float* __restrict__ left,
                         float* __restrict__ P) {
  int idx = blockIdx.x * blockDim.x + threadIdx.x;
  if (idx >= DR_ * R2_ * 1024) return;
  int d = idx / (R2_ * 1024);
  int rem = idx % (R2_ * 1024);
  int b2 = rem >> 10;
  int ij = rem & 1023;
  float s = 0.f;
  for (int a = 0; a < R1_; ++a) s += k3[(d * R1_ + a) * R2_ + b2] * left[a * 1024 + ij];
  P[idx] = s;
}

// ---------------- K5: W[d,ij,kl] = sum_b2 P[d,b2,ij] * right[b2,kl] ---------
// One wave (32 threads) per 16x16 output tile; fp32 WMMA, K=40 -> 10 steps.
__global__ void w_gemm_kernel(const float* __restrict__ P,
                              const float* __restrict__ right,
                              float* __restrict__ W) {
  const int lane = threadIdx.x;
  const int n0 = blockIdx.x * 16;   // kl
  const int m0 = blockIdx.y * 16;   // ij
  const int d  = blockIdx.z;
  const float* Pd = P + (size_t)d * (R2_ * 1024);
  float* Wd = W + (size_t)d * (1024 * 1024);
  const int lo   = lane & 15;
  const int rofs = (lane >> 4) * 8;
#if HAVE_WMMA_F32
  const int kb = (lane >> 4) * 2;
  v8f acc = {0.f,0.f,0.f,0.f,0.f,0.f,0.f,0.f};
  for (int kk = 0; kk < R2_; kk += 4) {
    v2f a, b;
    a.x = Pd[(size_t)(kk + kb    ) * 1024 + (m0 + lo)];
    a.y = Pd[(size_t)(kk + kb + 1) * 1024 + (m0 + lo)];
    b.x = right[(size_t)(kk + kb    ) * 1024 + (n0 + lo)];
    b.y = right[(size_t)(kk + kb + 1) * 1024 + (n0 + lo)];
    acc = __builtin_amdgcn_wmma_f32_16x16x4_f32(false, a, false, b, (short)0, acc,
                                                false, false);
  }
  for (int r = 0; r < 8; ++r)
    Wd[(size_t)(m0 + rofs + r) * 1024 + (n0 + lo)] = acc[r];
#else
  for (int r = 0; r < 8; ++r) {
    int m = m0 + rofs + r;
    float s = 0.f;
    for (int k = 0; k < R2_; ++k)
      s += Pd[(size_t)k * 1024 + m] * right[(size_t)k * 1024 + (n0 + lo)];
    Wd[(size_t)m * 1024 + (n0 + lo)] = s;
  }
#endif
}

// ---------------- K6: per-batch factored contraction -> W_out[b,32] --------
__global__ void wout_kernel(const float* __restrict__ rroot,
                            const float* __restrict__ ea,
                            const float* __restrict__ eb,
                            const float* __restrict__ ec,
                            const float* __restrict__ internal,
                            const float* __restrict__ left,
                            const float* __restrict__ right,
                            const int* __restrict__ miss,
                            float* __restrict__ wout) {
  int b = blockIdx.x;
  int t = threadIdx.x;                 // 128 threads
  __shared__ float rr[R3_], A_[32], B_[32], C_[32];
  __shared__ float T[R1_][R2_];
  __shared__ float V1[R2_];            // Rv (m<=2) or Lv (m>=3)
  __shared__ float G[R2_];             // G[a] or H[b2]
  if (t < R3_) rr[t] = rroot[b * R3_ + t];
  if (t < 32) { A_[t] = ea[b*32+t]; B_[t] = eb[b*32+t]; C_[t] = ec[b*32+t]; }
  __syncthreads();
  for (int e = t; e < R1_ * R2_; e += 128) {
    int a = e / R2_, b2 = e % R2_;
    float s = 0.f;
    for (int c = 0; c < R3_; ++c) s += rr[c] * internal[(c * R1_ + a) * R2_ + b2];
    T[a][b2] = s;
  }
  __syncthreads();
  int m = miss[0];
  if (m <= 2) {
    if (t < R2_) {                      // Rv[b2] = sum_{k,l} right[b2,k,l] eb[k] ec[l]
      const float* R = right + t * 1024;
      float s = 0.f;
      for (int k = 0; k < 32; ++k) {
        const float* Rk = R + k * 32;
        float sl = 0.f;
        for (int l = 0; l < 32; ++l) sl += Rk[l] * C_[l];
        s += B_[k] * sl;
      }
      V1[t] = s;
    }
    __syncthreads();
    if (t < R1_) {                      // G[a] = sum_b2 T[a][b2] Rv[b2]
      float s = 0.f;
      for (int b2 = 0; b2 < R2_; ++b2) s += T[t][b2] * V1[b2];
      G[t] = s;
    }
    __syncthreads();
    if (t < 32) {
      float s = 0.f;
      if (m == 1) {                     // out_i = sum_a G[a] * (left[a,i,:] . ea)
        for (int a = 0; a < R1_; ++a) {
          const float* L = left + (a * 32 + t) * 32;
          float sj = 0.f;
          for (int j = 0; j < 32; ++j) sj += L[j] * A_[j];
          s += G[a] * sj;
        }
      } else {                          // out_j = sum_a G[a] * (left[a,:,j] . ea)
        for (int a = 0; a < R1_; ++a) {
          const float* L = left + a * 1024 + t;
          float si = 0.f;
          for (int i = 0; i < 32; ++i) si += L[i * 32] * A_[i];
          s += G[a] * si;
        }
      }
      wout[b * 32 + t] = s;
    }
  } else {
    if (t < R1_) {                      // Lv[a] = sum_{i,j} left[a,i,j] ea[i] eb[j]
      const float* L = left + t * 1024;
      float s = 0.f;
      for (int i = 0; i < 32; ++i) {
        const float* Li = L + i * 32;
        float sj = 0.f;
        for (int j = 0; j < 32; ++j) sj += Li[j] * B_[j];
        s += A_[i] * sj;
      }
      V1[t] = s;
    }
    __syncthreads();
    if (t < R2_) {                      // H[b2] = sum_a T[a][b2] Lv[a]
      float s = 0.f;
      for (int a = 0; a < R1_; ++a) s += T[a][t] * V1[a];
      G[t] = s;
    }
    __syncthreads();
    if (t < 32) {
      float s = 0.f;
      if (m == 3) {                     // out_k = sum_b2 H[b2] * (right[b2,k,:] . ec)
        for (int b2 = 0; b2 < R2_; ++b2) {
          const float* R = right + (b2 * 32 + t) * 32;
          float sl = 0.f;
          for (int l = 0; l < 32; ++l) sl += R[l] * C_[l];
          s += G[b2] * sl;
        }
      } else {                          // out_l = sum_b2 H[b2] * (right[b2,:,l] . ec)
        for (int b2 = 0; b2 < R2_; ++b2) {
          const float* R = right + b2 * 1024 + t;
          float sk = 0.f;
          for (int k = 0; k < 32; ++k) sk += R[k * 32] * C_[k];
          s += G[b2] * sk;
        }
      }
      wout[b * 32 + t] = s;
    }
  }
}

// ---------------- K7: BatchNorm on W_out ------------------------------------
__global__ void bn_small_kernel(const float* __restrict__ x,
                                const float* __restrict__ gamma,
                                const float* __restrict__ beta,
                                float* __restrict__ y) {
  int col = threadIdx.x;               // 32
  float s = 0.f, ss = 0.f;
  for (int b = 0; b < BATCH; ++b) { float v = x[b*32+col]; s += v; ss += v*v; }
  float m = s / BATCH;
  float v = ss / BATCH - m * m;
  float inv = rsqrtf(v + EPS_);
  float g = gamma[col], bb = beta[col];
  for (int b = 0; b < BATCH; ++b) y[b*32+col] = (x[b*32+col] - m) * inv * g + bb;
}

// ---------------- K8: logits = W_out_bn @ E^T (fp32 WMMA, K=32) -------------
__global__ void logits_gemm_kernel(const float* __restrict__ A,   // [128][32]
                                   const float* __restrict__ E,   // [NENT][32]
                                   float* __restrict__ logits) {  // [128][NENT]
  const int lane = threadIdx.x;
  const int n0 = blockIdx.x * 16;      // entity tile
  const int m0 = blockIdx.y * 16;      // batch tile
  const int lo   = lane & 15;
  const int rofs = (lane >> 4) * 8;
#if HAVE_WMMA_F32
  const int kb = (lane >> 4) * 2;
  v8f acc = {0.f,0.f,0.f,0.f,0.f,0.f,0.f,0.f};
  for (int kk = 0; kk < 32; kk += 4) {
    v2f a, b;
    a.x = A[(size_t)(m0 + lo) * 32 + (kk + kb)];
    a.y = A[(size_t)(m0 + lo) * 32 + (kk + kb + 1)];
    b.x = E[(size_t)(n0 + lo) * 32 + (kk + kb)];
    b.y = E[(size_t)(n0 + lo) * 32 + (kk + kb + 1)];
    acc = __builtin_amdgcn_wmma_f32_16x16x4_f32(false, a, false, b, (short)0, acc,
                                                false, false);
  }
  for (int r = 0; r < 8; ++r)
    logits[(size_t)(m0 + rofs + r) * NENT + (n0 + lo)] = acc[r];
#else
  for (int r = 0; r < 8; ++r) {
    int mm = m0 + rofs + r;
    float s = 0.f;
    for (int k = 0; k < 32; ++k)
      s += A[(size_t)mm * 32 + k] * E[(size_t)(n0 + lo) * 32 + k];
    logits[(size_t)mm * NENT + (n0 + lo)] = s;
  }
#endif
}

// ---------------- K9: in-place row softmax ----------------------------------
__global__ void softmax_rows_kernel(float* __restrict__ pred) {
  int b = blockIdx.x;
  int t = threadIdx.x;                 // 256
  float* row = pred + (size_t)b * NENT;
  __shared__ float red[256];
  float mx = -3.4e38f;
  for (int i = t; i < NENT; i += 256) mx = fmaxf(mx, row[i]);
  red[t] = mx; __syncthreads();
  for (int s = 128; s > 0; s >>= 1) { if (t < s) red[t] = fmaxf(red[t], red[t+s]); __syncthreads(); }
  mx = red[0]; __syncthreads();
  float sum = 0.f;
  for (int i = t; i < NENT; i += 256) sum += expf(row[i] - mx);
  red[t] = sum; __syncthreads();
  for (int s = 128; s > 0; s >>= 1) { if (t < s) red[t] += red[t+s]; __syncthreads(); }
  float inv = 1.0f / red[0]; __syncthreads();
  for (int i = t; i < NENT; i += 256) row[i] = expf(row[i] - mx) * inv;
}

// ---------------- launcher ---------------------------------------------------
extern "C" void kernel_launch(void* const* d_in, const int* in_sizes, int n_in,
                              void* d_out, int out_size, void* d_ws, size_t ws_size,
                              hipStream_t stream) {
  const int*   r_idx   = (const int*)d_in[0];
  const int*   e2_idx  = (const int*)d_in[1];
  const int*   e3_idx  = (const int*)d_in[2];
  const int*   e4_idx  = (const int*)d_in[3];
  const int*   miss    = (const int*)d_in[4];
  const float* E       = (const float*)d_in[5];
  const float* Rw      = (const float*)d_in[6];
  const float* left    = (const float*)d_in[7];
  const float* right   = (const float*)d_in[8];
  const float* internal= (const float*)d_in[9];
  const float* root    = (const float*)d_in[10];
  const float* gr = (const float*)d_in[11];
  const float* br = (const float*)d_in[12];
  const float* ge = (const float*)d_in[13];
  const float* be = (const float*)d_in[14];
  const float* gw = (const float*)d_in[15];
  const float* bw = (const float*)d_in[16];

  float* out  = (float*)d_out;
  float* pred = out;                       // [128][100000]
  float* W    = out + PRED_ELEMS;          // [32][32^4]

  // workspace layout (floats)
  float* ws      = (float*)d_ws;
  float* remb    = ws;                     // 128*32
  float* ea      = ws + 4096;              // 128*32
  float* eb      = ws + 8192;
  float* ec      = ws + 12288;
  float* rroot   = ws + 16384;             // 128*40 = 5120
  float* k3buf   = ws + 21504;             // 32*40*40 = 51200
  float* pbuf    = ws + 72704;             // 32*40*1024 = 1310720
  float* woutbuf = ws + 1383424;           // 128*32
  float* woutbn  = ws + 1387520;           // 128*32

  bn_gather_kernel<<<4, 32, 0, stream>>>(Rw, E, r_idx, e2_idx, e3_idx, e4_idx,
                                         gr, br, ge, be, ws);
  rroot_kernel<<<(BATCH*R3_ + 255)/256, 256, 0, stream>>>(remb, root, rroot);
  k3_kernel<<<(DR_*R1_*R2_ + 255)/256, 256, 0, stream>>>(root, internal, k3buf);
  p_kernel<<<(DR_*R2_*1024 + 255)/256, 256, 0, stream>>>(k3buf, left, pbuf);
  w_gemm_kernel<<<dim3(64, 64, 32), 32, 0, stream>>>(pbuf, right, W);
  wout_kernel<<<BATCH, 128, 0, stream>>>(rroot, ea, eb, ec, internal, left, right,
                                         miss, woutbuf);
  bn_small_kernel<<<1, 32, 0, stream>>>(woutbuf, gw, bw, woutbn);
  logits_gemm_kernel<<<dim3(NENT/16, BATCH/16), 32, 0, stream>>>(woutbn, E, pred);
  softmax_rows_kernel<<<BATCH, 256, 0, stream>>>(pred);
}